// GraphTransformerBlock_21483426415183
// MI455X (gfx1250) — compile-verified
//
#include <hip/hip_runtime.h>
#include <hip/hip_bf16.h>
#include <math.h>

typedef float v2f __attribute__((ext_vector_type(2)));
typedef float v8f __attribute__((ext_vector_type(8)));

constexpr int NN = 100000;
constexpr int EE = 1600000;
constexpr int GG = 512;

// ---------------------------------------------------------------- zero init
__global__ void gtb_zero(float* __restrict__ p, size_t n) {
  size_t i = (size_t)blockIdx.x * blockDim.x + threadIdx.x;
  if (i < n) p[i] = 0.0f;
}

// ------------------------------------------------- fused node GEMM (q|k|v|skip)
// out[n, j] = sum_k x[n,k] * Wsel[j&63, k] + biassel[j&63],  j in [0,256)
__global__ __launch_bounds__(512) void gtb_node_gemm(
    const float* __restrict__ x,
    const float* __restrict__ Wq, const float* __restrict__ bq,
    const float* __restrict__ Wk, const float* __restrict__ bk,
    const float* __restrict__ Wv, const float* __restrict__ bv,
    const float* __restrict__ Ws, const float* __restrict__ bs,
    float* __restrict__ qkvs) {
  __shared__ float As[16 * 130];              // padded stride: conflict-free
  const int tid = threadIdx.x;
  const int m0  = blockIdx.x * 16;
  for (int i = tid; i < 16 * 128; i += 512)
    As[(i >> 7) * 130 + (i & 127)] = x[(size_t)(m0 + (i >> 7)) * 128 + (i & 127)];
  __syncthreads();

  const int wave = tid >> 5;
  const int lane = tid & 31;
  const int half = lane >> 4;
  const int n    = lane & 15;
  const int j    = wave * 16 + n;             // concat column 0..255
  const int sel  = j >> 6;
  const int row  = j & 63;
  const float* W = (sel == 0) ? Wq : (sel == 1) ? Wk : (sel == 2) ? Wv : Ws;
  const float* B = (sel == 0) ? bq : (sel == 1) ? bk : (sel == 2) ? bv : bs;
  const float bias = B[row];

  const float* arow = As + n * 130 + 2 * half;            // A[m=n][k0 + 2*half ..]
  const float* brow = W + (size_t)row * 128 + 2 * half;   // B[k][j] = W[j][k]

  v8f acc = {};
#pragma unroll
  for (int k0 = 0; k0 < 128; k0 += 4) {
    v2f a = *(const v2f*)(arow + k0);
    v2f b = *(const v2f*)(brow + k0);
    acc = __builtin_amdgcn_wmma_f32_16x16x4_f32(false, a, false, b,
                                                (short)0, acc, false, false);
  }
  float* out = qkvs + (size_t)m0 * 256 + j;
#pragma unroll
  for (int g = 0; g < 8; ++g)
    out[(size_t)(g + 8 * half) * 256] = acc[g] + bias;
}

// ------------------------------------------------------------- edge GEMM (e)
__global__ __launch_bounds__(128) void gtb_edge_gemm(
    const float* __restrict__ ea, const float* __restrict__ We,
    float* __restrict__ efeat) {
  __shared__ float As[16 * 34];
  const int tid = threadIdx.x;
  const int m0  = blockIdx.x * 16;
  for (int i = tid; i < 16 * 32; i += 128)
    As[(i >> 5) * 34 + (i & 31)] = ea[(size_t)(m0 + (i >> 5)) * 32 + (i & 31)];
  __syncthreads();

  const int wave = tid >> 5;
  const int lane = tid & 31;
  const int half = lane >> 4;
  const int n    = lane & 15;
  const int j    = wave * 16 + n;             // 0..63
  const float* arow = As + n * 34 + 2 * half;
  const float* brow = We + (size_t)j * 32 + 2 * half;

  v8f acc = {};
#pragma unroll
  for (int k0 = 0; k0 < 32; k0 += 4) {
    v2f a = *(const v2f*)(arow + k0);
    v2f b = *(const v2f*)(brow + k0);
    acc = __builtin_amdgcn_wmma_f32_16x16x4_f32(false, a, false, b,
                                                (short)0, acc, false, false);
  }
  float* out = efeat + (size_t)m0 * 64 + j;
#pragma unroll
  for (int g = 0; g < 8; ++g)
    out[(size_t)(g + 8 * half) * 64] = acc[g];
}

// -------------------------------------------- pass A: alpha + segment max
__device__ __forceinline__ unsigned f32key(float f) {
  unsigned u = __float_as_uint(f);
  return (u & 0x80000000u) ? ~u : (u | 0x80000000u);
}
__device__ __forceinline__ float keyf32(unsigned k) {
  return __uint_as_float((k & 0x80000000u) ? (k & 0x7FFFFFFFu) : ~k);
}

__global__ __launch_bounds__(256) void gtb_alpha(
    const int* __restrict__ eidx, const float* __restrict__ qkvs,
    const float* __restrict__ efeat, float* __restrict__ alpha,
    unsigned* __restrict__ mkeys) {
  const long long wv = (long long)blockIdx.x * 8 + (threadIdx.x >> 5);
  const long long e  = wv >> 1;
  if (e >= EE) return;
  const int h    = (int)(wv & 1);
  const int lane = threadIdx.x & 31;
  const int src  = eidx[e];
  const int dst  = eidx[EE + e];

  float qv = qkvs[(size_t)dst * 256 + h * 32 + lane];
  float kv = qkvs[(size_t)src * 256 + 64 + h * 32 + lane];
  float ev = efeat[(size_t)e * 64 + h * 32 + lane];
  float p  = qv * (kv + ev);
#pragma unroll
  for (int off = 16; off > 0; off >>= 1) p += __shfl_xor(p, off, 32);
  if (lane == 0) {
    float al = p * 0.17677669529663687f;    // 1/sqrt(32)
    alpha[e * 2 + h] = al;
    atomicMax(&mkeys[(size_t)dst * 2 + h], f32key(al));
  }
}

// ------------------- pass B: ex = exp(a-m); out_un += ex*v_e; den += ex
__global__ __launch_bounds__(256) void gtb_attn(
    const int* __restrict__ eidx, const float* __restrict__ qkvs,
    const float* __restrict__ efeat, const float* __restrict__ alpha,
    const unsigned* __restrict__ mkeys,
    float* __restrict__ outun, float* __restrict__ den) {
  const long long wv = (long long)blockIdx.x * 8 + (threadIdx.x >> 5);
  const long long e  = wv >> 1;
  if (e >= EE) return;
  const int h    = (int)(wv & 1);
  const int lane = threadIdx.x & 31;
  const int src  = eidx[e];
  const int dst  = eidx[EE + e];

  float al = alpha[e * 2 + h];
  float m  = keyf32(mkeys[(size_t)dst * 2 + h]);
  float ex = expf(al - m);
  float vv = qkvs[(size_t)src * 256 + 128 + h * 32 + lane]
           + efeat[(size_t)e * 64 + h * 32 + lane];
  atomicAdd(&outun[(size_t)dst * 64 + h * 32 + lane], ex * vv);
  if (lane == 0) atomicAdd(&den[(size_t)dst * 2 + h], ex);
}

// ---------------- node pass 1: normalize + skip; accumulate graph mean
__global__ __launch_bounds__(256) void gtb_node1(
    const float* __restrict__ outun, const float* __restrict__ den,
    const float* __restrict__ qkvs, const int* __restrict__ batch,
    float* __restrict__ val, float* __restrict__ gsum, float* __restrict__ gcnt) {
  const size_t i = (size_t)blockIdx.x * 256 + threadIdx.x;
  const int n = (int)(i >> 6), hc = (int)(i & 63), h = hc >> 5;
  float v = outun[i] / (den[(size_t)n * 2 + h] + 1e-16f)
          + qkvs[(size_t)n * 256 + 192 + hc];
  val[i] = v;
  const int g = batch[n];
  atomicAdd(&gsum[g * 64 + hc], v);
  if (hc == 0) atomicAdd(&gcnt[g], 1.0f);
}

__global__ void gtb_gmean(float* __restrict__ gsum, const float* __restrict__ gcnt) {
  const int i = blockIdx.x * 256 + threadIdx.x;     // G*64
  const float c = gcnt[i >> 6];
  gsum[i] = (c > 0.0f) ? gsum[i] / c : 0.0f;
}

// ---------------- node pass 2: center; accumulate graph var
__global__ __launch_bounds__(256) void gtb_node2(
    const int* __restrict__ batch, const float* __restrict__ mscale,
    const float* __restrict__ gmean, float* __restrict__ val,
    float* __restrict__ gvar) {
  const size_t i = (size_t)blockIdx.x * 256 + threadIdx.x;
  const int n = (int)(i >> 6), hc = (int)(i & 63);
  const int g = batch[n];
  float c = val[i] - mscale[hc] * gmean[g * 64 + hc];
  val[i] = c;
  atomicAdd(&gvar[g * 64 + hc], c * c);
}

__global__ void gtb_gistd(float* __restrict__ gvar, const float* __restrict__ gcnt) {
  const int i = blockIdx.x * 256 + threadIdx.x;     // G*64
  const float c = gcnt[i >> 6];
  const float var = (c > 0.0f) ? gvar[i] / c : 0.0f;
  gvar[i] = 1.0f / sqrtf(var + 1e-5f);
}

// ---------------- node pass 3: scale, shift, relu
__global__ __launch_bounds__(256) void gtb_node3(
    const int* __restrict__ batch, const float* __restrict__ gistd,
    const float* __restrict__ gw, const float* __restrict__ gb,
    float* __restrict__ val) {
  const size_t i = (size_t)blockIdx.x * 256 + threadIdx.x;
  const int n = (int)(i >> 6), hc = (int)(i & 63);
  const int g = batch[n];
  float o = val[i] * gistd[g * 64 + hc] * gw[hc] + gb[hc];
  val[i] = fmaxf(o, 0.0f);
}

extern "C" void kernel_launch(void* const* d_in, const int* in_sizes, int n_in,
                              void* d_out, int out_size, void* d_ws, size_t ws_size,
                              hipStream_t stream) {
  const float* x     = (const float*)d_in[0];
  const int*   eidx  = (const int*)d_in[1];
  const int*   batch = (const int*)d_in[2];
  const float* eattr = (const float*)d_in[3];
  const float* Wq = (const float*)d_in[4];
  const float* bq = (const float*)d_in[5];
  const float* Wk = (const float*)d_in[6];
  const float* bk = (const float*)d_in[7];
  const float* Wv = (const float*)d_in[8];
  const float* bv = (const float*)d_in[9];
  const float* We = (const float*)d_in[10];
  const float* Ws = (const float*)d_in[11];
  const float* bs = (const float*)d_in[12];
  const float* gw = (const float*)d_in[13];
  const float* gb = (const float*)d_in[14];
  const float* gm = (const float*)d_in[15];
  float* out = (float*)d_out;

  // workspace layout (floats)
  float* qkvs  = (float*)d_ws;                       // N*256
  float* efeat = qkvs + (size_t)NN * 256;            // E*64
  float* alpha = efeat + (size_t)EE * 64;            // E*2
  float* zbase = alpha + (size_t)EE * 2;             // ---- zeroed region ----
  unsigned* mkeys = (unsigned*)zbase;                // N*2
  float* den   = zbase + (size_t)NN * 2;             // N*2
  float* outun = den + (size_t)NN * 2;               // N*64
  float* gsum  = outun + (size_t)NN * 64;            // G*64
  float* gcnt  = gsum + GG * 64;                     // G
  float* gvar  = gcnt + GG;                          // G*64
  const size_t zcount = (size_t)NN * 2 * 2 + (size_t)NN * 64 + (size_t)GG * 129;

  gtb_zero<<<(int)((zcount + 255) / 256), 256, 0, stream>>>(zbase, zcount);
  gtb_node_gemm<<<NN / 16, 512, 0, stream>>>(x, Wq, bq, Wk, bk, Wv, bv, Ws, bs, qkvs);
  gtb_edge_gemm<<<EE / 16, 128, 0, stream>>>(eattr, We, efeat);
  gtb_alpha<<<(2 * EE) / 8, 256, 0, stream>>>(eidx, qkvs, efeat, alpha, mkeys);
  gtb_attn<<<(2 * EE) / 8, 256, 0, stream>>>(eidx, qkvs, efeat, alpha, mkeys, outun, den);
  gtb_node1<<<(NN * 64) / 256, 256, 0, stream>>>(outun, den, qkvs, batch, out, gsum, gcnt);
  gtb_gmean<<<(GG * 64) / 256, 256, 0, stream>>>(gsum, gcnt);
  gtb_node2<<<(NN * 64) / 256, 256, 0, stream>>>(batch, gm, gsum, out, gvar);
  gtb_gistd<<<(GG * 64) / 256, 256, 0, stream>>>(gvar, gcnt);
  gtb_node3<<<(NN * 64) / 256, 256, 0, stream>>>(batch, gvar, gw, gb, out);
}